// SAGEmodel_70351564308951
// MI455X (gfx1250) — compile-verified
//
#include <hip/hip_runtime.h>
#include <stdint.h>

// GraphSAGE 2-layer forward for MI455X (gfx1250, wave32).
// Aggregation: L2-resident f32 atomics (one wave per edge, float4 per lane).
// GEMMs: V_WMMA_F32_16X16X4_F32 (full fp32 accuracy; GEMM is ~6.5 GFLOP,
// far below the memory/atomic cost, so low-precision WMMA buys nothing).
// Dropout: exact JAX threefry2x32-20 reimplementation, key (0,42).

#define N_NODES 50000
#define D_FEAT  128
#define N_EDGES 800000

typedef __attribute__((ext_vector_type(2))) float v2f;
typedef __attribute__((ext_vector_type(8))) float v8f;

// ---------------------------------------------------------------- zero fill
__global__ void zero_f32(float* __restrict__ p, int n) {
    int i = blockIdx.x * blockDim.x + threadIdx.x;
    int stride = gridDim.x * blockDim.x;
    for (; i < n; i += stride) p[i] = 0.0f;
}

// --------------------------------------------------------- edge scatter-add
// One wave32 per edge: lane L moves float4 at column 4*L (128 floats total).
__global__ __launch_bounds__(256)
void scatter_mean(const float* __restrict__ feat,
                  const long long* __restrict__ src,
                  const long long* __restrict__ dst,
                  float* __restrict__ agg,
                  float* __restrict__ cnt,
                  int doCount) {
    int wave   = (blockIdx.x * blockDim.x + threadIdx.x) >> 5;
    int lane   = threadIdx.x & 31;
    int nwaves = (gridDim.x * blockDim.x) >> 5;
    for (int e = wave; e < N_EDGES; e += nwaves) {
        int s = (int)src[e];
        int d = (int)dst[e];
        const float4 v = ((const float4*)(feat + (size_t)s * D_FEAT))[lane];
        float* o = agg + (size_t)d * D_FEAT + lane * 4;
        atomicAdd(o + 0, v.x);
        atomicAdd(o + 1, v.y);
        atomicAdd(o + 2, v.z);
        atomicAdd(o + 3, v.w);
        if (doCount && lane == 0) atomicAdd(cnt + d, 1.0f);
    }
}

// ------------------------------------------------- JAX threefry2x32 dropout
__device__ __forceinline__ unsigned rotl32(unsigned v, int r) {
    return (v << r) | (v >> (32 - r));
}

__device__ __forceinline__ void threefry2x32(unsigned x0, unsigned x1,
                                             unsigned& o0, unsigned& o1) {
    // key = jax.random.key(42) -> (k0=0, k1=42)
    const unsigned ks[3] = {0u, 42u, 0u ^ 42u ^ 0x1BD11BDAu};
    const int R0[4] = {13, 15, 26, 6};
    const int R1[4] = {17, 29, 16, 24};
    x0 += ks[0];
    x1 += ks[1];
#pragma unroll
    for (int g = 0; g < 5; ++g) {
        const int* R = (g & 1) ? R1 : R0;
#pragma unroll
        for (int i = 0; i < 4; ++i) {
            x0 += x1;
            x1 = rotl32(x1, R[i]);
            x1 ^= x0;
        }
        x0 += ks[(g + 1) % 3];
        x1 += ks[(g + 2) % 3] + (unsigned)(g + 1);
    }
    o0 = x0;
    o1 = x1;
}

__device__ __forceinline__ float dropout_keep(unsigned e, float val) {
    // jax random_bits for even count n: counters (i, i+n/2); out0 -> first
    // half, out1 -> second half. uniform = bitcast((bits>>9)|0x3f800000)-1.
    const unsigned HALF = (N_NODES * D_FEAT) / 2;  // 3200000
    unsigned o0, o1, bits;
    if (e < HALF) {
        threefry2x32(e, e + HALF, o0, o1);
        bits = o0;
    } else {
        threefry2x32(e - HALF, e, o0, o1);
        bits = o1;
    }
    float u = __uint_as_float((bits >> 9) | 0x3f800000u) - 1.0f;
    return (u < 0.8f) ? val * 1.25f : 0.0f;  // keep -> /(1-p), p=0.2
}

// ------------------------------------------------------------ fused SAGE GEMM
// out_tile[16 x 128] = mean_tile[16x128] @ Wl^T + x_tile[16x128] @ Wr^T + b
// Block = 256 threads = 8 waves; wave w owns output columns [16w, 16w+16).
// K marched in steps of 4 via V_WMMA_F32_16X16X4_F32.
template <int LAYER>
__launch_bounds__(256) __global__
void sage_gemm(const float* __restrict__ agg,
               const float* __restrict__ cnt,
               const float* __restrict__ selfF,
               const float* __restrict__ Wl,
               const float* __restrict__ Wr,
               const float* __restrict__ bias,
               float* __restrict__ out) {
    __shared__ float Am[16 * D_FEAT];  // mean rows
    __shared__ float Ax[16 * D_FEAT];  // self rows
    const int row0 = blockIdx.x * 16;

    for (int idx = threadIdx.x; idx < 16 * D_FEAT; idx += 256) {
        int r = idx >> 7, c = idx & 127;
        int g = row0 + r;
        float inv = 1.0f / fmaxf(cnt[g], 1.0f);
        Am[idx] = agg[(size_t)g * D_FEAT + c] * inv;
        Ax[idx] = selfF[(size_t)g * D_FEAT + c];
    }
    __syncthreads();

    const int wv   = threadIdx.x >> 5;
    const int lane = threadIdx.x & 31;
    const int m    = lane & 15;             // A row (M) held by this lane
    const int kb   = (lane >> 4) << 1;      // k sub-offset: 0 (lo half) / 2 (hi)
    const int ncol = wv * 16 + (lane & 15); // global output column (B's N)

    v8f acc = {};
#pragma unroll 4
    for (int k = 0; k < D_FEAT; k += 4) {
        v2f am, bl, ax, br;
        // A 16x4 f32 layout: lanes 0-15 hold K=kb..kb+1, lanes 16-31 K=2..3
        am.x = Am[m * D_FEAT + k + kb];
        am.y = Am[m * D_FEAT + k + kb + 1];
        // B 4x16: B[k][n] = Wl[n][k] (out = A @ Wl^T), mirrored K split
        bl.x = Wl[(size_t)ncol * D_FEAT + k + kb];
        bl.y = Wl[(size_t)ncol * D_FEAT + k + kb + 1];
        acc = __builtin_amdgcn_wmma_f32_16x16x4_f32(
            false, am, false, bl, (short)0, acc, false, false);

        ax.x = Ax[m * D_FEAT + k + kb];
        ax.y = Ax[m * D_FEAT + k + kb + 1];
        br.x = Wr[(size_t)ncol * D_FEAT + k + kb];
        br.y = Wr[(size_t)ncol * D_FEAT + k + kb + 1];
        acc = __builtin_amdgcn_wmma_f32_16x16x4_f32(
            false, ax, false, br, (short)0, acc, false, false);
    }

    // D 16x16 f32: VGPR v -> M = v + 8*lane[4]; N = lane[3:0]
    const float bv    = bias[ncol];
    const int   mbase = (lane >> 4) << 3;
#pragma unroll
    for (int v = 0; v < 8; ++v) {
        int   gr  = row0 + mbase + v;
        float val = acc[v] + bv;
        if (LAYER == 1) {
            val = fmaxf(val, 0.0f);
            val = dropout_keep((unsigned)(gr * D_FEAT + ncol), val);
        }
        out[(size_t)gr * D_FEAT + ncol] = val;
    }
}

// --------------------------------------------------------------------- host
extern "C" void kernel_launch(void* const* d_in, const int* in_sizes, int n_in,
                              void* d_out, int out_size, void* d_ws, size_t ws_size,
                              hipStream_t stream) {
    const float*     x   = (const float*)d_in[0];
    const long long* ei  = (const long long*)d_in[1];  // int64 [2, 800000]
    const float*     W1l = (const float*)d_in[2];
    const float*     W1r = (const float*)d_in[3];
    const float*     b1  = (const float*)d_in[4];
    const float*     W2l = (const float*)d_in[5];
    const float*     W2r = (const float*)d_in[6];
    const float*     b2  = (const float*)d_in[7];
    float*           out = (float*)d_out;

    // Workspace partition (reuse agg between layers): ~51.4 MB total.
    float* agg = (float*)d_ws;
    float* cnt = agg + (size_t)N_NODES * D_FEAT;
    float* h   = cnt + N_NODES;

    const long long* src = ei;
    const long long* dst = ei + N_EDGES;

    // Layer 1
    zero_f32<<<2048, 256, 0, stream>>>(agg, N_NODES * D_FEAT + N_NODES);
    scatter_mean<<<4096, 256, 0, stream>>>(x, src, dst, agg, cnt, 1);
    sage_gemm<1><<<N_NODES / 16, 256, 0, stream>>>(agg, cnt, x, W1l, W1r, b1, h);

    // Layer 2 (degree counts identical; recompute only agg)
    zero_f32<<<2048, 256, 0, stream>>>(agg, N_NODES * D_FEAT);
    scatter_mean<<<4096, 256, 0, stream>>>(h, src, dst, agg, cnt, 0);
    sage_gemm<2><<<N_NODES / 16, 256, 0, stream>>>(agg, cnt, h, W2l, W2r, b2, out);
}